// Network_77489799954988
// MI455X (gfx1250) — compile-verified
//
#include <hip/hip_runtime.h>
#include <math.h>

// ---- problem dims ----
#define NB   64
#define NN   1024
#define WDIM 64
#define RR   4
#define IND  512
#define HID  512
#define OUTD 256
#define CATD 768          // IND + RR*WDIM
#define XID  471          // WD*R + 3*WD + 5*R + 3

// output (floats) offsets: output | xi | rw_new | memory_new | usage_new
#define O_OUT  0
#define O_XI   16384      // 64*256
#define O_RW   46528      // + 64*471
#define O_MEM  308672     // + 64*1024*4
#define O_USE  4502976    // + 64*1024*64

// workspace (floats) offsets
#define W_RF   0          // read_flat [64][256]
#define W_H    16384      // h [64][512]
#define W_WW   49152      // ww [64][1024]
#define W_P    114688     // P [64][4]
#define W_Q    114944     // Q [64][4]
#define W_FWD  115200     // fwd_w [64][1024][4]
#define W_BWD  377344     // bwd_w [64][1024][4]

typedef __attribute__((ext_vector_type(2))) float v2f;
typedef __attribute__((ext_vector_type(8))) float v8f;
typedef __attribute__((ext_vector_type(4))) unsigned int u32x4;
typedef __attribute__((ext_vector_type(4))) int i32x4;
typedef __attribute__((ext_vector_type(8))) int i32x8;

__device__ __forceinline__ v8f wmma4(v2f a, v2f b, v8f c) {
  // D = A(16x4 f32) * B(4x16 f32) + C(16x16 f32); fp32 tensor path, no downcast
  return __builtin_amdgcn_wmma_f32_16x16x4_f32(false, a, false, b, (short)0, c,
                                               false, false);
}

__device__ __forceinline__ float sigmoidf_(float x) {
  return 1.f / (1.f + __expf(-x));
}
__device__ __forceinline__ float oneplusf_(float x) {
  return 1.f + (x > 20.f ? x : log1pf(__expf(x)));
}

// ---------------- 1) read_flat[b, r*64+w] = sum_n M[b,n,w] * rw[b,n,r] -------
__global__ void k_read_flat(const float* __restrict__ mem,
                            const float* __restrict__ rw,
                            float* __restrict__ rf) {
  int b = blockIdx.x, t = threadIdx.x;
  int r = t >> 6, w = t & 63;
  const float* M = mem + (size_t)b * NN * WDIM;
  const float* R = rw + (size_t)b * NN * RR;
  float acc = 0.f;
  for (int n = 0; n < NN; ++n) {
    if ((n & 255) == 0 && n + 256 < NN)
      __builtin_prefetch(&M[(n + 256) * WDIM + w], 0, 1);  // global_prefetch_b8
    acc += M[n * WDIM + w] * R[n * RR + r];
  }
  rf[b * (RR * WDIM) + r * WDIM + w] = acc;
}

// ---------------- 2) h = relu([x|rf] @ W1 + b1)  (WMMA f32 16x16x4) ----------
__device__ __forceinline__ float catld(const float* __restrict__ x,
                                       const float* __restrict__ rf,
                                       int m, int k) {
  return (k < IND) ? x[m * IND + k] : rf[m * (RR * WDIM) + (k - IND)];
}

__global__ void k_gemm_h(const float* __restrict__ x, const float* __restrict__ rf,
                         const float* __restrict__ W1, const float* __restrict__ b1,
                         float* __restrict__ h) {
  int nt = blockIdx.x, mt = blockIdx.y, lane = threadIdx.x;
  int half = lane >> 4, cx = lane & 15;
  int n = nt * 16 + cx;
  int mrow = mt * 16 + cx;
  v8f c = {0.f, 0.f, 0.f, 0.f, 0.f, 0.f, 0.f, 0.f};
  for (int k = 0; k < CATD; k += 4) {
    int kk = k + half * 2;
    v2f a, bf;
    a.x = catld(x, rf, mrow, kk);
    a.y = catld(x, rf, mrow, kk + 1);
    bf.x = W1[kk * HID + n];
    bf.y = W1[(kk + 1) * HID + n];
    c = wmma4(a, bf, c);
  }
  float bias = b1[n];
#pragma unroll
  for (int v = 0; v < 8; ++v) {
    int m = mt * 16 + v + half * 8;
    h[m * HID + n] = fmaxf(c[v] + bias, 0.f);
  }
}

// ---------------- 3) output = h@Wo + rf@Wr + bo + br -------------------------
__global__ void k_gemm_out(const float* __restrict__ h, const float* __restrict__ rf,
                           const float* __restrict__ Wo, const float* __restrict__ bo,
                           const float* __restrict__ Wr, const float* __restrict__ br,
                           float* __restrict__ out) {
  int nt = blockIdx.x, mt = blockIdx.y, lane = threadIdx.x;
  int half = lane >> 4, cx = lane & 15;
  int n = nt * 16 + cx;
  int mrow = mt * 16 + cx;
  v8f c = {0.f, 0.f, 0.f, 0.f, 0.f, 0.f, 0.f, 0.f};
  for (int k = 0; k < HID; k += 4) {
    int kk = k + half * 2;
    v2f a, bf;
    a.x = h[mrow * HID + kk];
    a.y = h[mrow * HID + kk + 1];
    bf.x = Wo[kk * OUTD + n];
    bf.y = Wo[(kk + 1) * OUTD + n];
    c = wmma4(a, bf, c);
  }
  for (int k = 0; k < RR * WDIM; k += 4) {
    int kk = k + half * 2;
    v2f a, bf;
    a.x = rf[mrow * (RR * WDIM) + kk];
    a.y = rf[mrow * (RR * WDIM) + kk + 1];
    bf.x = Wr[kk * OUTD + n];
    bf.y = Wr[(kk + 1) * OUTD + n];
    c = wmma4(a, bf, c);
  }
  float bias = bo[n] + br[n];
#pragma unroll
  for (int v = 0; v < 8; ++v) {
    int m = mt * 16 + v + half * 8;
    out[m * OUTD + n] = c[v] + bias;
  }
}

// ---------------- 4) xi = h @ Wxi + bxi  (471 cols, padded to 480) -----------
__global__ void k_gemm_xi(const float* __restrict__ h, const float* __restrict__ Wxi,
                          const float* __restrict__ bxi, float* __restrict__ xi) {
  int nt = blockIdx.x, mt = blockIdx.y, lane = threadIdx.x;
  int half = lane >> 4, cx = lane & 15;
  int n = nt * 16 + cx;
  bool ok = (n < XID);
  int mrow = mt * 16 + cx;
  v8f c = {0.f, 0.f, 0.f, 0.f, 0.f, 0.f, 0.f, 0.f};
  for (int k = 0; k < HID; k += 4) {
    int kk = k + half * 2;
    v2f a, bf;
    a.x = h[mrow * HID + kk];
    a.y = h[mrow * HID + kk + 1];
    bf.x = ok ? Wxi[kk * XID + n] : 0.f;
    bf.y = ok ? Wxi[(kk + 1) * XID + n] : 0.f;
    c = wmma4(a, bf, c);
  }
  float bias = ok ? bxi[n] : 0.f;
#pragma unroll
  for (int v = 0; v < 8; ++v) {
    int m = mt * 16 + v + half * 8;
    if (ok) xi[m * XID + n] = c[v] + bias;
  }
}

// -------- 5) interface: usage_new, allocation (bitonic sort), write cw, ww ---
__global__ void __launch_bounds__(1024)
k_interface(const float* __restrict__ xi_all, const float* __restrict__ mem,
            const float* __restrict__ rw, const float* __restrict__ usage,
            const float* __restrict__ wwp, const float* __restrict__ prec,
            float* __restrict__ usage_out, float* __restrict__ ww_ws,
            float* __restrict__ Pws, float* __restrict__ Qws) {
  __shared__ float s_key[WDIM];
  __shared__ float s_free[RR];
  __shared__ float s_scal[4];  // strength, ag, wg, inv key norm
  __shared__ float s_sortv[NN];
  __shared__ int s_sorti[NN];
  __shared__ float s_wc[NN];
  __shared__ float s_alloc[NN];
  __shared__ float s_red[NN];
  __shared__ float s_pq[8];
  int b = blockIdx.x, t = threadIdx.x;
  const float* xi = xi_all + b * XID;
  if (t < WDIM) s_key[t] = xi[260 + t];
  if (t < RR) s_free[t] = sigmoidf_(xi[453 + t]);
  if (t == 64) s_scal[0] = oneplusf_(xi[324]);  // write strength
  if (t == 65) s_scal[1] = sigmoidf_(xi[457]);  // ag
  if (t == 66) s_scal[2] = sigmoidf_(xi[458]);  // wg
  if (t < 8) s_pq[t] = 0.f;
  __syncthreads();
  if (t == 0) {
    float s = 0.f;
    for (int w = 0; w < WDIM; ++w) s += s_key[w] * s_key[w];
    s_scal[3] = rsqrtf(s + 1e-6f);
  }
  __syncthreads();
  // usage_new + sort payload
  const float* rwb = rw + (size_t)b * NN * RR;
  float ret = 1.f;
  for (int r = 0; r < RR; ++r) ret *= (1.f - s_free[r] * rwb[t * RR + r]);
  float u = usage[b * NN + t], wp = wwp[b * NN + t];
  float un = (u + wp - u * wp) * ret;
  usage_out[b * NN + t] = un;
  s_sortv[t] = un;
  s_sorti[t] = t;
  // write content similarity
  const float* mr = mem + ((size_t)b * NN + t) * WDIM;
  float msq = 0.f, dot = 0.f;
  for (int w = 0; w < WDIM; ++w) {
    float m = mr[w];
    msq += m * m;
    dot += m * s_key[w];
  }
  s_wc[t] = dot * rsqrtf(msq + 1e-6f) * s_scal[3] * s_scal[0];
  __syncthreads();
  // softmax over N
  s_red[t] = s_wc[t];
  __syncthreads();
  for (int s = 512; s > 0; s >>= 1) {
    if (t < s) s_red[t] = fmaxf(s_red[t], s_red[t + s]);
    __syncthreads();
  }
  float mx = s_red[0];
  __syncthreads();
  float e = __expf(s_wc[t] - mx);
  s_red[t] = e;
  __syncthreads();
  for (int s = 512; s > 0; s >>= 1) {
    if (t < s) s_red[t] += s_red[t + s];
    __syncthreads();
  }
  float inv = 1.f / s_red[0];
  __syncthreads();
  s_wc[t] = e * inv;
  // bitonic argsort ascending on usage_new
  for (int k = 2; k <= NN; k <<= 1)
    for (int j = k >> 1; j > 0; j >>= 1) {
      __syncthreads();
      int ixj = t ^ j;
      if (ixj > t) {
        float v1 = s_sortv[t], v2 = s_sortv[ixj];
        bool up = ((t & k) == 0);
        if (up ? (v1 > v2) : (v1 < v2)) {
          s_sortv[t] = v2;
          s_sortv[ixj] = v1;
          int tmp = s_sorti[t];
          s_sorti[t] = s_sorti[ixj];
          s_sorti[ixj] = tmp;
        }
      }
    }
  __syncthreads();
  // inclusive cumprod scan on sorted usage
  s_red[t] = s_sortv[t];
  __syncthreads();
  for (int off = 1; off < NN; off <<= 1) {
    float tmp = (t >= off) ? s_red[t - off] : 1.f;
    __syncthreads();
    s_red[t] *= tmp;
    __syncthreads();
  }
  float cpe = (t == 0) ? 1.f : s_red[t - 1];
  s_alloc[s_sorti[t]] = (1.f - s_sortv[t]) * cpe;
  __syncthreads();
  float ag = s_scal[1], wg = s_scal[2];
  float wwn = wg * (ag * s_alloc[t] + (1.f - ag) * s_wc[t]);
  ww_ws[b * NN + t] = wwn;
  float pr = prec[b * NN + t];
  for (int r = 0; r < RR; ++r) {
    float rv = rwb[t * RR + r];
    atomicAdd(&s_pq[r], wwn * rv);      // Q_r = sum_j ww_j rw_jr (ds_add_f32)
    atomicAdd(&s_pq[4 + r], pr * rv);   // P_r = sum_j prec_j rw_jr
  }
  __syncthreads();
  if (t < RR) {
    Qws[b * RR + t] = s_pq[t];
    Pws[b * RR + t] = s_pq[4 + t];
  }
}

// ---------------- 6) memory_new (elementwise erase/write) --------------------
__global__ void k_memwrite(const float* __restrict__ mem,
                           const float* __restrict__ xi_all,
                           const float* __restrict__ ww,
                           float* __restrict__ memout) {
  size_t gid = (size_t)blockIdx.x * blockDim.x + threadIdx.x;
  int w = (int)(gid & 63);
  size_t ng = gid >> 6;  // b*N + n
  int b = (int)(ng >> 10);
  const float* xi = xi_all + b * XID;
  float e = sigmoidf_(xi[325 + w]);
  float wv = xi[389 + w];
  float wwv = ww[ng];
  memout[gid] = mem[gid] * (1.f - wwv * e) + wwv * wv;
}

// ---- 7) link matvecs via fp32 WMMA without materializing link_new ----------
// C1 = L_tile(16x4) x B, C2 = L^T_tile(16x4) x B over full K=1024,
// B columns: [rw(4) | ww*rw(4) | zeros(8)]
// Link tiles are staged into LDS by the Tensor Data Mover (double-buffered,
// TENSORcnt-tracked); row tile uses TDM pad feature (1 DWORD / 128 DWORDs)
// to give a 129-float row stride and avoid 16-way LDS bank conflicts.

#define LCH 128                 // K-chunk length
#define LNC (NN / LCH)          // 8 chunks
#define ROWP (LCH + 1)          // padded row stride in LDS

#if __has_builtin(__builtin_amdgcn_tensor_load_to_lds)
#define HAVE_TDM 1
#else
#define HAVE_TDM 0
#endif

#if HAVE_TDM
// 2D TDM load: tile_y rows of tile_x fp32 elements, row stride = stride_elems.
__device__ __forceinline__ void tdm_load_2d(unsigned lds_addr, const void* gaddr,
                                            unsigned tile_x, unsigned tile_y,
                                            unsigned stride_elems, bool pad_row) {
  unsigned long long ga = (unsigned long long)(uintptr_t)gaddr;
  u32x4 g0 = {1u,                                   // count=1 (valid user D#)
              lds_addr,                              // LDS byte address
              (unsigned)(ga & 0xFFFFFFFFu),          // global_addr[31:0]
              (unsigned)((ga >> 32) & 0x01FFFFFFu)   // global_addr[56:32]
                  | (2u << 30)};                     // type=2 ("image")
  unsigned w0 = (2u << 16);                          // data_size=4B
  if (pad_row) w0 |= (1u << 20) | (6u << 22);        // pad_en, every 128 DW, +1 DW
  i32x8 g1 = {(int)w0,
              (int)((tile_x & 0xFFFFu) << 16),                              // tdim0 lo
              (int)(((tile_x >> 16) & 0xFFFFu) | ((tile_y & 0xFFFFu) << 16)),  // tdim0 hi | tdim1 lo
              (int)(((tile_y >> 16) & 0xFFFFu) | ((tile_x & 0xFFFFu) << 16)),  // tdim1 hi | tile_dim0
              (int)(tile_y & 0xFFFFu),                                      // tile_dim1
              (int)stride_elems,                                            // dim0_stride lo
              0, 0};
  i32x4 z4 = {0, 0, 0, 0};
#if __clang_major__ >= 23
  i32x8 z8 = {0, 0, 0, 0, 0, 0, 0, 0};
  __builtin_amdgcn_tensor_load_to_lds(g0, g1, z4, z4, z8, 0);
#else
  __builtin_amdgcn_tensor_load_to_lds(g0, g1, z4, z4, 0);
#endif
}
#endif

__global__ void k_link(const float* __restrict__ link, const float* __restrict__ rw,
                       const float* __restrict__ ww, const float* __restrict__ prec,
                       const float* __restrict__ Pws, const float* __restrict__ Qws,
                       float* __restrict__ fwd, float* __restrict__ bwd) {
  int b = blockIdx.y;
  int i0 = blockIdx.x * 16;
  int lane = threadIdx.x;
  int half = lane >> 4, cx = lane & 15;
  const float* Lb = link + (size_t)b * NN * NN;
  const float* rwb = rw + (size_t)b * NN * RR;
  const float* wwb = ww + b * NN;
  v8f c1 = {0.f, 0.f, 0.f, 0.f, 0.f, 0.f, 0.f, 0.f};
  v8f c2 = {0.f, 0.f, 0.f, 0.f, 0.f, 0.f, 0.f, 0.f};

#if HAVE_TDM
  __shared__ float s_row[2][16 * ROWP];  // row tile, padded stride 129
  __shared__ float s_col[2][LCH * 16];   // transposed-source tile [j][i]
  // issue chunk 0
  tdm_load_2d((unsigned)(uintptr_t)&s_row[0][0], &Lb[(size_t)i0 * NN + 0],
              LCH, 16, NN, true);
  tdm_load_2d((unsigned)(uintptr_t)&s_col[0][0], &Lb[(size_t)0 * NN + i0],
              16, LCH, NN, false);
  for (int cchunk = 0; cchunk < LNC; ++cchunk) {
    int buf = cchunk & 1;
    if (cchunk + 1 < LNC) {
      int j0n = (cchunk + 1) * LCH;
      tdm_load_2d((unsigned)(uintptr_t)&s_row[buf ^ 1][0],
                  &Lb[(size_t)i0 * NN + j0n], LCH, 16, NN, true);
      tdm_load_2d((unsigned)(uintptr_t)&s_col[buf ^ 1][0],
                  &Lb[(size_t)j0n * NN + i0], 16, LCH, NN, false);
      __builtin_amdgcn_s_wait_tensorcnt(2);  // current chunk's 2 loads done
    } else {
      __builtin_amdgcn_s_wait_tensorcnt(0);
    }
    int j0 = cchunk * LCH;
    for (int jj = 0; jj < LCH; jj += 4) {
      int kk = jj + half * 2;
      v2f a1, a2, bf;
      a1.x = s_row[buf][cx * ROWP + kk];
      a1.y = s_row[buf][cx * ROWP + kk + 1];
      a2.x = s_col[buf][kk * 16 + cx];
      a2.y = s_col[buf][(kk + 1) * 16 + cx];
      int jg = j0 + kk;
      float b0 = 0.f, b1v = 0.f;
      if (cx < 8) {
        int r = cx & 3;
        float r0 = rwb[jg * RR + r], r1 = rwb[(jg + 1) * RR + r];
        if (cx < 4) {
          b0 = r0;
          b1v = r1;
        } else {
          b0 = r0 * wwb[jg];
          b1v = r1 * wwb[jg + 1];
        }
      }
      bf.x = b0;
      bf.y = b1v;
      c1 = wmma4(a1, bf, c1);
      c2 = wmma4(a2, bf, c2);
    }
  }
#else
  for (int j = 0; j < NN; j += 4) {
    int kk = j + half * 2;
    v2f a1, a2, bf;
    a1.x = Lb[(size_t)(i0 + cx) * NN + kk];
    a1.y = Lb[(size_t)(i0 + cx) * NN + kk + 1];
    a2.x = Lb[(size_t)kk * NN + i0 + cx];
    a2.y = Lb[(size_t)(kk + 1) * NN + i0 + cx];
    float b0 = 0.f, b1v = 0.f;
    if (cx < 8) {
      int r = cx & 3;
      float r0 = rwb[kk * RR + r], r1 = rwb[(kk + 1) * RR + r];
      if (cx < 4) {
        b0 = r0;
        b1v = r1;
      } else {
        b0 = r0 * wwb[kk];
        b1v = r1 * wwb[kk + 1];
      }
    }
    bf.x = b0;
    bf.y = b1v;
    c1 = wmma4(a1, bf, c1);
    c2 = wmma4(a2, bf, c2);
  }
#endif

  // combine: fwd = (1-ww_i)S1 - S2 + ww_i P_r - diag ;
  //          bwd = (1-ww_i)T1 - T2 + prec_i Q_r - diag
#pragma unroll
  for (int v = 0; v < 8; ++v) {
    float s1 = c1[v], t1 = c2[v];
    float s2 = __shfl_xor(s1, 4, 32);  // col r+4 -> col r (same half)
    float t2 = __shfl_xor(t1, 4, 32);
    if (cx < 4) {
      int r = cx;
      int i = i0 + v + half * 8;
      float wwi = wwb[i];
      float pri = prec[b * NN + i];
      float lii = Lb[(size_t)i * NN + i];
      float rwir = rwb[i * RR + r];
      float diag = ((1.f - 2.f * wwi) * lii + wwi * pri) * rwir;
      float Pr = Pws[b * RR + r], Qr = Qws[b * RR + r];
      size_t o = ((size_t)b * NN + i) * RR + r;
      fwd[o] = (1.f - wwi) * s1 - s2 + wwi * Pr - diag;
      bwd[o] = (1.f - wwi) * t1 - t2 + pri * Qr - diag;
    }
  }
}

// ----- 8) read content weights on memory_new + assemble rw_new ---------------
__global__ void __launch_bounds__(1024)
k_read_rw(const float* __restrict__ memnew, const float* __restrict__ xi_all,
          const float* __restrict__ fwd, const float* __restrict__ bwd,
          float* __restrict__ rw_out) {
  __shared__ float s_k[RR][WDIM];
  __shared__ float s_invn[RR], s_str[RR];
  __shared__ float s_modes[RR][3];
  __shared__ float s_sim[RR][NN];
  __shared__ float s_red[NN];
  int b = blockIdx.x, t = threadIdx.x;
  const float* xi = xi_all + b * XID;
  if (t < RR * WDIM) s_k[t >> 6][t & 63] = xi[t];
  if (t >= 256 && t < 256 + RR) {
    int r = t - 256;
    s_str[r] = oneplusf_(xi[256 + r]);
  }
  if (t >= 260 && t < 260 + RR) {
    int r = t - 260;
    float m0 = xi[459 + r * 3], m1 = xi[459 + r * 3 + 1], m2 = xi[459 + r * 3 + 2];
    float mm = fmaxf(m0, fmaxf(m1, m2));
    float e0 = __expf(m0 - mm), e1 = __expf(m1 - mm), e2 = __expf(m2 - mm);
    float inv = 1.f / (e0 + e1 + e2);
    s_modes[r][0] = e0 * inv;
    s_modes[r][1] = e1 * inv;
    s_modes[r][2] = e2 * inv;
  }
  __syncthreads();
  if (t < RR) {
    float s = 0.f;
    for (int w = 0; w < WDIM; ++w) s += s_k[t][w] * s_k[t][w];
    s_invn[t] = rsqrtf(s + 1e-6f);
  }
  __syncthreads();
  const float* mr = memnew + ((size_t)b * NN + t) * WDIM;
  float msq = 0.f, d0 = 0.f, d1 = 0.f, d2 = 0.f, d3 = 0.f;
  for (int w = 0; w < WDIM; ++w) {
    float m = mr[w];
    msq += m * m;
    d0 += m * s_k[0][w];
    d1 += m * s_k[1][w];
    d2 += m * s_k[2][w];
    d3 += m * s_k[3][w];
  }
  float im = rsqrtf(msq + 1e-6f);
  s_sim[0][t] = d0 * im * s_invn[0] * s_str[0];
  s_sim[1][t] = d1 * im * s_invn[1] * s_str[1];
  s_sim[2][t] = d2 * im * s_invn[2] * s_str[2];
  s_sim[3][t] = d3 * im * s_invn[3] * s_str[3];
  __syncthreads();
  for (int r = 0; r < RR; ++r) {
    s_red[t] = s_sim[r][t];
    __syncthreads();
    for (int s = 512; s > 0; s >>= 1) {
      if (t < s) s_red[t] = fmaxf(s_red[t], s_red[t + s]);
      __syncthreads();
    }
    float mx = s_red[0];
    __syncthreads();
    float e = __expf(s_sim[r][t] - mx);
    s_red[t] = e;
    __syncthreads();
    for (int s = 512; s > 0; s >>= 1) {
      if (t < s) s_red[t] += s_red[t + s];
      __syncthreads();
    }
    float inv = 1.f / s_red[0];
    __syncthreads();
    s_sim[r][t] = e * inv;
    __syncthreads();
  }
  size_t base = ((size_t)b * NN + t) * RR;
  for (int r = 0; r < RR; ++r) {
    rw_out[base + r] = s_modes[r][0] * bwd[base + r] +
                       s_modes[r][1] * s_sim[r][t] +
                       s_modes[r][2] * fwd[base + r];
  }
}

extern "C" void kernel_launch(void* const* d_in, const int* in_sizes, int n_in,
                              void* d_out, int out_size, void* d_ws, size_t ws_size,
                              hipStream_t stream) {
  const float* x = (const float*)d_in[0];
  const float* rw = (const float*)d_in[1];
  const float* mem = (const float*)d_in[2];
  const float* usage = (const float*)d_in[3];
  const float* wwp = (const float*)d_in[4];
  const float* link = (const float*)d_in[5];
  const float* prec = (const float*)d_in[6];
  const float* W1 = (const float*)d_in[7];
  const float* b1 = (const float*)d_in[8];
  const float* Wo = (const float*)d_in[9];
  const float* bo = (const float*)d_in[10];
  const float* Wr = (const float*)d_in[11];
  const float* br = (const float*)d_in[12];
  const float* Wxi = (const float*)d_in[13];
  const float* bxi = (const float*)d_in[14];

  float* out = (float*)d_out;
  float* ws = (float*)d_ws;
  float* rf = ws + W_RF;
  float* h = ws + W_H;
  float* ww = ws + W_WW;
  float* P = ws + W_P;
  float* Q = ws + W_Q;
  float* fw = ws + W_FWD;
  float* bw = ws + W_BWD;
  float* o_out = out + O_OUT;
  float* o_xi = out + O_XI;
  float* o_rw = out + O_RW;
  float* o_mem = out + O_MEM;
  float* o_use = out + O_USE;

  k_read_flat<<<NB, 256, 0, stream>>>(mem, rw, rf);
  k_gemm_h<<<dim3(HID / 16, NB / 16), 32, 0, stream>>>(x, rf, W1, b1, h);
  k_gemm_out<<<dim3(OUTD / 16, NB / 16), 32, 0, stream>>>(h, rf, Wo, bo, Wr, br, o_out);
  k_gemm_xi<<<dim3((XID + 15) / 16, NB / 16), 32, 0, stream>>>(h, Wxi, bxi, o_xi);
  k_interface<<<NB, 1024, 0, stream>>>(o_xi, mem, rw, usage, wwp, prec, o_use, ww, P, Q);
  k_memwrite<<<(NB * NN * WDIM) / 256, 256, 0, stream>>>(mem, o_xi, ww, o_mem);
  k_link<<<dim3(NN / 16, NB), 32, 0, stream>>>(link, rw, ww, prec, P, Q, fw, bw);
  k_read_rw<<<NB, 1024, 0, stream>>>(o_mem, o_xi, fw, bw, o_rw);
}